// SingleMofifModel_81415400063421
// MI455X (gfx1250) — compile-verified
//
#include <hip/hip_runtime.h>

// Problem constants (must match the reference)
constexpr int Bn = 8192;
constexpr int Ln = 2048;
constexpr int W1n = 24;
constexpr int ROWS_PER_BLOCK = 16;           // one wave (wave32) per row
constexpr int NBLK = Bn / ROWS_PER_BLOCK;    // 512 blocks
constexpr int THREADS = 32 * ROWS_PER_BLOCK; // 512 threads = 16 waves

typedef __attribute__((ext_vector_type(2))) float v2f;
typedef __attribute__((ext_vector_type(8))) float v8f;

__device__ inline float waveReduceF(float v) {
#pragma unroll
  for (int m = 16; m > 0; m >>= 1) v += __shfl_xor(v, m, 32);
  return v;
}

// One block = 16 rows of X; wave w handles row b = blockIdx.x*16 + w.
// Emits one f32 partial per block into `partials` (no atomics -> deterministic).
__global__ __launch_bounds__(THREADS) void motif_loglik_main(
    const float* __restrict__ phi,    // (4,4,4)
    const float* __restrict__ pwm,    // (4,24,4)
    const float* __restrict__ gamma,  // (B,4)
    const int*   __restrict__ dw,     // (4,)
    const int*   __restrict__ Zp,     // (B,)
    const int*   __restrict__ X,      // (B,L)
    float*       __restrict__ partials) {
  // --- LDS log-LUTs: kills all transcendentals in the hot loop ---
  __shared__ float lut3[64];    // log phi[i,j,k]
  __shared__ float lut1[16];    // log( mean_i phi[i,j,k] )
  __shared__ float lut0[4];     // log( mean_{i,j} phi[i,j,k] )
  __shared__ float lutp[384];   // log pwm[r,pos,base]
  __shared__ float sS[ROWS_PER_BLOCK];
  __shared__ float sSc[ROWS_PER_BLOCK][4];

  const int tid = threadIdx.x;
  if (tid < 64) {
    lut3[tid] = logf(phi[tid]);
  } else if (tid < 80) {
    const int jk = tid - 64;
    lut1[jk] = logf(0.25f * (phi[jk] + phi[16 + jk] + phi[32 + jk] + phi[48 + jk]));
  } else if (tid < 84) {
    const int k = tid - 80;
    float s = 0.f;
#pragma unroll
    for (int m = 0; m < 16; ++m) s += phi[m * 4 + k];
    lut0[k] = logf(s * (1.0f / 16.0f));
  } else if (tid >= 96 && tid < 480) {
    const int m = tid - 96;
    lutp[m] = logf(pwm[m]);
  }
  __syncthreads();

  const int wid  = tid >> 5;     // wave id = row-in-block
  const int lane = tid & 31;
  const int b    = blockIdx.x * ROWS_PER_BLOCK + wid;
  const int* __restrict__ Xrow = X + (size_t)b * Ln;
  const int4* __restrict__ Xrow4 = reinterpret_cast<const int4*>(Xrow);

  // --- S_bg[b] = sum_l log(bg[b,l]); vectorized int4 path, l = 4..2047 ---
  float acc = 0.f;
  for (int q = 1 + lane; q < (Ln / 4); q += 32) {
    const int4 v = Xrow4[q];                 // l = 4q .. 4q+3 (coalesced b128)
    const int xm1 = Xrow[4 * q - 1];         // cache-resident rereads
    const int xm2 = Xrow[4 * q - 2];
    acc += lut3[xm2 * 16 + xm1 * 4 + v.x];
    acc += lut3[xm1 * 16 + v.x * 4 + v.y];
    acc += lut3[v.x * 16 + v.y * 4 + v.z];
    acc += lut3[v.y * 16 + v.z * 4 + v.w];
  }
  if (lane == 0) {  // l = 0,1,2,3
    const int x0 = Xrow[0], x1 = Xrow[1], x2 = Xrow[2], x3 = Xrow[3];
    acc += lut0[x0] + lut1[x0 * 4 + x1];
    acc += lut3[x0 * 16 + x1 * 4 + x2];
    acc += lut3[x1 * 16 + x2 * 4 + x3];
  }
  const float S = waveReduceF(acc);

  // --- motif correction: lane = offset within motif window ---
  const int Zb = Zp[b];
  const int4 wv = *reinterpret_cast<const int4*>(dw);
  float t0 = 0.f, t1 = 0.f, t2 = 0.f, t3 = 0.f;
  if (lane < W1n) {
    const int l  = Zb + lane;
    const int xv = Xrow[l];
    float logbg;
    if (l == 0)      logbg = lut0[xv];
    else if (l == 1) logbg = lut1[Xrow[0] * 4 + xv];
    else             logbg = lut3[Xrow[l - 2] * 16 + Xrow[l - 1] * 4 + xv];
    const int oi = lane * 4 + xv;
    t0 = (lane < wv.x) ? (lutp[oi]       - logbg) : 0.f;
    t1 = (lane < wv.y) ? (lutp[96 + oi]  - logbg) : 0.f;
    t2 = (lane < wv.z) ? (lutp[192 + oi] - logbg) : 0.f;
    t3 = (lane < wv.w) ? (lutp[288 + oi] - logbg) : 0.f;
  }
  const float sc0 = waveReduceF(t0);
  const float sc1 = waveReduceF(t1);
  const float sc2 = waveReduceF(t2);
  const float sc3 = waveReduceF(t3);

  if (lane == 0) {
    sS[wid] = S;
    sSc[wid][0] = sc0; sSc[wid][1] = sc1; sSc[wid][2] = sc2; sSc[wid][3] = sc3;
  }
  __syncthreads();

  // --- wave 0: Frobenius combine via V_WMMA_F32_16X16X4_F32 ---
  // A (16x4): A[i][r] = S[i] + score[i][r]
  //   lanes 0-15 : vgpr0 = K0, vgpr1 = K1 ; lanes 16-31: vgpr0 = K2, vgpr1 = K3
  // B (4x16): B[k][j] = gamma[row j][k]  (same half-wave K split)
  // D[i][j] = sum_r A[i][r]*gamma[j][r]  -> trace(D) = block contribution
  if (tid < 32) {                     // uniform per wave; EXEC all-1 inside
    const int i    = lane & 15;
    const int half = lane >> 4;       // 0 -> r=0,1 ; 1 -> r=2,3
    const float Si = sS[i];
    v2f a, bm;
    a.x = Si + sSc[i][half * 2 + 0];
    a.y = Si + sSc[i][half * 2 + 1];
    const int brow = blockIdx.x * ROWS_PER_BLOCK + i;
    bm.x = gamma[brow * 4 + half * 2 + 0];
    bm.y = gamma[brow * 4 + half * 2 + 1];
    v8f c = {};
    c = __builtin_amdgcn_wmma_f32_16x16x4_f32(
        /*neg_a=*/false, a, /*neg_b=*/false, bm,
        /*c_mod=*/(short)0, c, /*reuse_a=*/false, /*reuse_b=*/false);
    // trace extraction from documented C/D layout
    float diag = 0.f;
    if (lane < 8)        diag = c[lane];
    else if (lane >= 24) diag = c[lane - 24];
    const float tr = waveReduceF(diag);
    if (lane == 0) partials[blockIdx.x] = tr;
  }
}

// Deterministic fixed-order reduction of the 512 per-block partials.
__global__ __launch_bounds__(NBLK) void motif_loglik_reduce(
    const float* __restrict__ partials, float* __restrict__ out) {
  __shared__ float s[NBLK];
  const int t = threadIdx.x;
  s[t] = partials[t];
  __syncthreads();
  for (int h = NBLK / 2; h > 0; h >>= 1) {
    if (t < h) s[t] += s[t + h];
    __syncthreads();
  }
  if (t == 0) out[0] = s[0];
}

extern "C" void kernel_launch(void* const* d_in, const int* in_sizes, int n_in,
                              void* d_out, int out_size, void* d_ws, size_t ws_size,
                              hipStream_t stream) {
  const float* phi   = (const float*)d_in[0];
  const float* pwm   = (const float*)d_in[1];
  const float* gamma = (const float*)d_in[2];
  const int*   w     = (const int*)d_in[3];
  const int*   Z     = (const int*)d_in[4];
  const int*   X     = (const int*)d_in[5];
  float* out      = (float*)d_out;
  float* partials = (float*)d_ws;   // NBLK * sizeof(float) = 2 KB

  motif_loglik_main<<<NBLK, THREADS, 0, stream>>>(phi, pwm, gamma, w, Z, X, partials);
  motif_loglik_reduce<<<1, NBLK, 0, stream>>>(partials, out);
}